// LieNet_5866925326753
// MI455X (gfx1250) — compile-verified
//
#include <hip/hip_runtime.h>
#include <hip/hip_bf16.h>
#include <math.h>

typedef __attribute__((ext_vector_type(2))) float v2f;
typedef __attribute__((ext_vector_type(8))) float v8f;

#define NB   128
#define S2   171      // S0/2 joints after space pool
#define F0C  100      // input frames
#define F2C  25       // frames after first time pool
#define GOUT 7        // groups after second time pool (ceil(25/4))
#define KDIM 4788     // 171*7*4
#define OD   20

// ---------- small 3x3 helpers ----------
__device__ __forceinline__ void mm3(const float* __restrict__ A,
                                    const float* __restrict__ B,
                                    float* __restrict__ C) {
#pragma unroll
  for (int a = 0; a < 3; ++a)
#pragma unroll
    for (int c = 0; c < 3; ++c)
      C[a * 3 + c] = A[a * 3 + 0] * B[0 * 3 + c] +
                     A[a * 3 + 1] * B[1 * 3 + c] +
                     A[a * 3 + 2] * B[2 * 3 + c];
}

__device__ __forceinline__ float ang3(const float* __restrict__ M) {
  float c = 0.5f * (M[0] + M[4] + M[8] - 1.0f);
  c = fminf(fmaxf(c, -1.0f + 1e-6f), 1.0f - 1e-6f);
  return acosf(c);
}

__device__ __forceinline__ void cpy9(float* __restrict__ d, const float* __restrict__ s) {
#pragma unroll
  for (int i = 0; i < 9; ++i) d[i] = s[i];
}

// x[n,a,b,s,f] with S=342, F=100  -> load 3x3 matrix for fixed (n,s,f)
__device__ __forceinline__ void load_x(const float* __restrict__ x, int n, int s, int f,
                                       float* __restrict__ M) {
  const int base = n * 9 * (342 * F0C) + s * F0C + f;
#pragma unroll
  for (int e = 0; e < 9; ++e) M[e] = x[base + e * (342 * F0C)];
}

// ---------------------------------------------------------------------------
// Kernel 1: fully fused rotmap(w1)+pool_space -> rotmap(w2)+pool_time ->
//           rotmap(w3)+pool_time -> logmap + clip.  Writes yT [128, 4788].
// One thread per (n, z, g).  Reads each element of x exactly once overall,
// so this kernel runs at the HBM roofline (~157 MB @ 23.3 TB/s ~= 7 us).
// ---------------------------------------------------------------------------
__global__ __launch_bounds__(256) void lienet_fused_kernel(
    const float* __restrict__ x, const float* __restrict__ w1,
    const float* __restrict__ w2, const float* __restrict__ w3,
    float* __restrict__ yT) {
  const int total = NB * S2 * GOUT;
  int tid = blockIdx.x * blockDim.x + threadIdx.x;
  if (tid >= total) return;

  const int g = tid % GOUT;
  const int z = (tid / GOUT) % S2;
  const int n = tid / (GOUT * S2);

  float W1a[9], W1b[9], W2[9], W3[9];
  cpy9(W1a, w1 + (2 * z) * 9);
  cpy9(W1b, w1 + (2 * z + 1) * 9);
  cpy9(W2, w2 + z * 9);
  cpy9(W3, w3 + z * 9);

  float best3[9];
  float bestA3 = -2.0f;

#pragma unroll
  for (int j = 0; j < 4; ++j) {
    const int f2 = 4 * g + j;
    if (f2 >= F2C) break;  // padded frames in last group never win (ang=-1 in ref)

    float best2[9];
    float bestA2 = -2.0f;
#pragma unroll
    for (int i = 0; i < 4; ++i) {
      const int f1 = 4 * f2 + i;  // always < 100
      float X0[9], X1[9], A0[9], A1[9];
      load_x(x, n, 2 * z, f1, X0);
      load_x(x, n, 2 * z + 1, f1, X1);
      mm3(W1a, X0, A0);
      mm3(W1b, X1, A1);
      const float aa0 = ang3(A0);
      const float aa1 = ang3(A1);
      const float* P = (aa1 > aa0) ? A1 : A0;  // argmax picks first on tie
      float Q[9];
      mm3(W2, P, Q);
      const float aq = ang3(Q);
      if (aq > bestA2) { bestA2 = aq; cpy9(best2, Q); }
    }
    float R[9];
    mm3(W3, best2, R);
    const float ar = ang3(R);
    if (ar > bestA3) { bestA3 = ar; cpy9(best3, R); }
  }

  // log-map:  fi = normalize([m21-m12, m20-m02, m10-m01]), angle a
  const float a = bestA3;
  const float sc = a / (2.0f * sinf(a));
  float l0 = sc * (best3[7] - best3[5]);
  float l1 = sc * (best3[6] - best3[2]);
  float l2 = sc * (best3[3] - best3[1]);
  const float inv = 1.0f / sqrtf(l0 * l0 + l1 * l1 + l2 * l2);
  float y0 = l0 * inv, y1 = l1 * inv, y2 = l2 * inv, y3 = a;

  // relu-eps clip: |y|<0.3 -> sign-snapped 0.3 (y==0 -> +0.3, per reference)
  y0 = (fabsf(y0) < 0.3f) ? ((y0 < 0.0f) ? -0.3f : 0.3f) : y0;
  y1 = (fabsf(y1) < 0.3f) ? ((y1 < 0.0f) ? -0.3f : 0.3f) : y1;
  y2 = (fabsf(y2) < 0.3f) ? ((y2 < 0.0f) ? -0.3f : 0.3f) : y2;
  y3 = (fabsf(y3) < 0.3f) ? ((y3 < 0.0f) ? -0.3f : 0.3f) : y3;

  float* dst = yT + n * KDIM + (z * GOUT + g) * 4;
  dst[0] = y0; dst[1] = y1; dst[2] = y2; dst[3] = y3;
}

// ---------------------------------------------------------------------------
// Kernel 2: out[n,o] = sum_k yT[n,k] * fc_w[o,k] + fc_b[o]   via
// V_WMMA_F32_16X16X4_F32.  Grid (8,2) tiles of 16x16; 8 waves/block split
// K = 1197 wmma-steps (~150 each), and each wave runs TWO independent
// accumulator chains (even/odd steps) so the serial D->C dependency depth
// drops to ~75 wmma, letting loads and wmma of the two chains interleave.
// A frag: lane l(0..15)=row M, vgpr pair holds K = half*2 + {0,1}. B mirrored.
// ---------------------------------------------------------------------------
__global__ __launch_bounds__(256) void lienet_fc_wmma_kernel(
    const float* __restrict__ yT, const float* __restrict__ fcw,
    const float* __restrict__ fcb, float* __restrict__ out) {
  __shared__ float red[8][32][8];

  const int lane = threadIdx.x & 31;
  const int wave = threadIdx.x >> 5;
  const int half = lane >> 4;
  const int l = lane & 15;
  const int mt = blockIdx.x;  // 0..7  (rows n)
  const int nt = blockIdx.y;  // 0..1  (cols o, padded to 32)

  const int row = mt * 16 + l;
  const int col = nt * 16 + l;
  const float bmask = (col < OD) ? 1.0f : 0.0f;  // mask (no divergent branch:
  const int colc = (col < OD) ? col : (OD - 1);  //  EXEC stays all-1s at wmma)

  const float* __restrict__ arow = yT + row * KDIM + half * 2;
  const float* __restrict__ brow = fcw + colc * KDIM + half * 2;

  const int KSTEPS = KDIM / 4;          // 1197
  const int CH = (KSTEPS + 7) / 8;      // 150
  const int kbeg = wave * CH;
  const int kend = (kbeg + CH < KSTEPS) ? (kbeg + CH) : KSTEPS;

  v8f acc0 = {};
  v8f acc1 = {};
  int ks = kbeg;
  for (; ks + 1 < kend; ks += 2) {
    const int k0 = ks * 4;
    const int k1 = k0 + 4;
    v2f A0, B0, A1, B1;
    A0.x = arow[k0];     A0.y = arow[k0 + 1];
    B0.x = bmask * brow[k0];     B0.y = bmask * brow[k0 + 1];
    A1.x = arow[k1];     A1.y = arow[k1 + 1];
    B1.x = bmask * brow[k1];     B1.y = bmask * brow[k1 + 1];
    acc0 = __builtin_amdgcn_wmma_f32_16x16x4_f32(
        false, A0, false, B0, (short)0, acc0, false, false);
    acc1 = __builtin_amdgcn_wmma_f32_16x16x4_f32(
        false, A1, false, B1, (short)0, acc1, false, false);
  }
  if (ks < kend) {  // odd remainder (only last wave: 147 steps)
    const int k = ks * 4;
    v2f A, B;
    A.x = arow[k];
    A.y = arow[k + 1];
    B.x = bmask * brow[k];
    B.y = bmask * brow[k + 1];
    acc0 = __builtin_amdgcn_wmma_f32_16x16x4_f32(
        false, A, false, B, (short)0, acc0, false, false);
  }

#pragma unroll
  for (int r = 0; r < 8; ++r) red[wave][lane][r] = acc0[r] + acc1[r];
  __syncthreads();

  if (wave == 0) {
#pragma unroll
    for (int r = 0; r < 8; ++r) {
      float v = 0.0f;
#pragma unroll
      for (int w = 0; w < 8; ++w) v += red[w][lane][r];
      // C/D layout: vgpr r, lanes 0-15 -> M=r, lanes 16-31 -> M=r+8; N=lane%16
      const int m = mt * 16 + half * 8 + r;
      const int o = nt * 16 + l;
      if (o < OD) out[m * OD + o] = v + fcb[o];
    }
  }
}

extern "C" void kernel_launch(void* const* d_in, const int* in_sizes, int n_in,
                              void* d_out, int out_size, void* d_ws, size_t ws_size,
                              hipStream_t stream) {
  const float* x   = (const float*)d_in[0];  // [128,3,3,342,100]
  const float* w1  = (const float*)d_in[1];  // [342,3,3]
  const float* w2  = (const float*)d_in[2];  // [171,3,3]
  const float* w3  = (const float*)d_in[3];  // [171,3,3]
  const float* fcw = (const float*)d_in[4];  // [20,4788]
  const float* fcb = (const float*)d_in[5];  // [20]
  float* out = (float*)d_out;                // [128,20]
  float* yT = (float*)d_ws;                  // [128,4788] scratch (~2.45 MB)

  const int total = NB * S2 * GOUT;  // 153216
  lienet_fused_kernel<<<(total + 255) / 256, 256, 0, stream>>>(x, w1, w2, w3, yT);
  lienet_fc_wmma_kernel<<<dim3(8, 2), 256, 0, stream>>>(yT, fcw, fcb, out);
}